// DifferentiableStack_24902220382576
// MI455X (gfx1250) — compile-verified
//
#include <hip/hip_runtime.h>

// DifferentiableStack on MI455X (gfx1250).
//
// Exact algebraic reduction: out_t = stack[:,63,:] after update, and slot 63
// evolves as y_t = a_t*y_{t-1} + g_t*x_t with a_t=(1-o)(1-p), g_t=p(1-o)
// (stack data only flows away from slot 63). First-order linear scan.
//
// Pipeline:
//   K0: gate precompute a[b][t], g[b][t]
//   K1: chunk summaries (zero-init run) -> Bfin, Atot          (x -> L2)
//   K2: 16-chunk sequential scan -> exclusive Init
//   K3: per (b,chunk,channel-half) recompute with carry; 16-timestep blocks as
//       OUT(16x16) = Atilde(16x20) @ Btilde(20x16) via 5x V_WMMA_F32_16X16X4_F32.
//       g folded into Atilde; Btilde = [carry row; 16 raw-x rows (async-staged
//       via GLOBAL_LOAD_ASYNC_TO_LDS_B128, double-buffered); 3 zero pad rows]
//       -> fully uniform (branchless) fragment loads.
//
// Roofline: ~256 MB HBM traffic -> ~11 us at 23.3 TB/s; x (128 MB) resident in
// 192 MB L2 between K1 and K3.

typedef __attribute__((ext_vector_type(2))) float v2f;
typedef __attribute__((ext_vector_type(8))) float v8f;
typedef __attribute__((ext_vector_type(4))) int   v4i;
typedef __attribute__((address_space(1))) v4i gv4i;   // global <4 x i32>
typedef __attribute__((address_space(3))) v4i lv4i;   // LDS    <4 x i32>

#define B_  16
#define L_  4096
#define D_  512
#define NC  16          // chunks along L
#define S_  (L_ / NC)   // 256 timesteps per chunk
#define TB  16          // timestep tile = WMMA M
#define KP  20          // padded K: 1 carry + 16 steps + 3 pad
#define DB  256         // channels per K3 block (2 blocks per (b,chunk))

#if defined(__gfx1250__) && __has_builtin(__builtin_amdgcn_global_load_async_to_lds_b128)
#define HAS_ASYNC_LDS 1
#else
#define HAS_ASYNC_LDS 0
#endif

#if __has_builtin(__builtin_amdgcn_s_wait_asynccnt)
#define WAIT_ASYNC(n) __builtin_amdgcn_s_wait_asynccnt(n)
#elif defined(__gfx1250__)
#define WAIT_ASYNC(n) asm volatile("s_wait_asynccnt %0" :: "i"(n) : "memory")
#else
#define WAIT_ASYNC(n)
#endif

// ---------------------------------------------------------------- K0: gates
__global__ void k0_gates(const float* __restrict__ pg, const float* __restrict__ og,
                         float* __restrict__ a, float* __restrict__ g) {
    int i = blockIdx.x * blockDim.x + threadIdx.x;     // over B_*L_
    if (i < B_ * L_) {
        float p = pg[i], o = og[i];
        float om = 1.0f - o;
        a[i] = (1.0f - p) * om;
        g[i] = p * om;
    }
}

// ------------------------------------------------------- K1: chunk summaries
__global__ __launch_bounds__(256) void k1_chunks(
    const float* __restrict__ x, const float* __restrict__ a,
    const float* __restrict__ g, float* __restrict__ Bfin,
    float* __restrict__ Atot) {
    int d  = (blockIdx.x % (D_ / 256)) * 256 + threadIdx.x;
    int bc = blockIdx.x / (D_ / 256);
    int b  = bc / NC, c = bc % NC;
    const float* xp = x + ((size_t)b * L_ + (size_t)c * S_) * D_ + d;
    const float* ap = a + b * L_ + c * S_;
    const float* gp = g + b * L_ + c * S_;
    float y = 0.0f, A = 1.0f;
    for (int s = 0; s < S_; ++s) {
        if (s + 64 < S_)
            __builtin_prefetch(xp + (size_t)(s + 64) * D_, 0, 1); // global_prefetch_b8
        float as = ap[s];
        y = fmaf(as, y, gp[s] * xp[(size_t)s * D_]);
        A *= as;
    }
    Bfin[(size_t)bc * D_ + d] = y;
    if (d == 0) Atot[bc] = A;
}

// ------------------------------------------------- K2: scan over chunk level
__global__ void k2_scan(const float* __restrict__ Bfin, const float* __restrict__ Atot,
                        float* __restrict__ Init) {
    int i = blockIdx.x * blockDim.x + threadIdx.x;     // over B_*D_
    if (i >= B_ * D_) return;
    int b = i / D_, d = i % D_;
    float y = 0.0f;
    for (int c = 0; c < NC; ++c) {
        size_t idx = (size_t)(b * NC + c) * D_ + d;
        Init[idx] = y;                                  // exclusive prefix
        y = fmaf(Atot[b * NC + c], y, Bfin[idx]);
    }
}

#if HAS_ASYNC_LDS
// async-copy one 16 x DB x-tile (16 KB): 4 b128 issues per thread (ASYNCcnt +4)
__device__ __forceinline__ void stage_tile(const float* __restrict__ src_rows,
                                           float* dst_rows, int tid) {
#pragma unroll
    for (int j = 0; j < 4; ++j) {
        int flat = tid + 256 * j;          // units of 16 bytes
        int row  = flat >> 6;              // DB floats = 64 x 16B per row
        int coff = (flat & 63) << 2;       // float offset within row
        const float* src = src_rows + (size_t)row * D_ + coff;
        float*       dst = dst_rows + row * DB + coff;
        __builtin_amdgcn_global_load_async_to_lds_b128(
            (gv4i*)(unsigned long long)src,
            (lv4i*)(unsigned int)(unsigned long long)dst,   // low 32 bits = LDS offset
            0, 0);
    }
}
#else
__device__ __forceinline__ void stage_tile(const float* __restrict__ src_rows,
                                           float* dst_rows, int tid) {
#pragma unroll
    for (int j = 0; j < 4; ++j) {
        int flat = tid + 256 * j;
        int row  = flat >> 6;
        int coff = (flat & 63) << 2;
        float4 v = *(const float4*)(src_rows + (size_t)row * D_ + coff);
        *(float4*)(dst_rows + row * DB + coff) = v;
    }
}
#endif

// ---------------------------------------------- K3: WMMA-driven output pass
__global__ __launch_bounds__(256) void k3_out(
    const float* __restrict__ x, const float* __restrict__ a,
    const float* __restrict__ g, const float* __restrict__ Init,
    float* __restrict__ out) {
    __shared__ float sA[TB * KP];        // Atilde (g folded in): 16 x 20
    __shared__ float sBT[2][KP * DB];    // Btilde double buffered: 20 rows x DB
                                         //   row 0 = carry, 1..16 = x, 17..19 = 0

    int dblk = blockIdx.x & 1;
    int bc   = blockIdx.x >> 1;
    int b = bc / NC, c = bc % NC;
    int dbase = dblk * DB;
    int tid  = threadIdx.x;
    int lane = tid & 31;
    int wave = tid >> 5;
    int row  = lane & 15;
    int half = lane >> 4;

    const float* ap = a + b * L_ + c * S_;
    const float* gp = g + b * L_ + c * S_;
    const float* xp = x + ((size_t)b * L_ + (size_t)c * S_) * D_ + dbase;
    float*       op = out + ((size_t)b * L_ + (size_t)c * S_) * D_ + dbase;

    // zero pad rows 17..19 of both buffers; carry row 0 of buf0 = Init
    for (int i = tid; i < 3 * DB; i += 256) {
        sBT[0][17 * DB + i] = 0.0f;
        sBT[1][17 * DB + i] = 0.0f;
    }
    for (int dd = tid; dd < DB; dd += 256)
        sBT[0][dd] = Init[(size_t)bc * D_ + dbase + dd];

    stage_tile(xp, &sBT[0][DB], tid);    // prologue: x rows of sub-block 0
    int cur = 0;

    for (int sb = 0; sb < S_ / TB; ++sb) {
        int t0 = sb * TB;
        bool more = (sb + 1 < S_ / TB);
        if (more) stage_tile(xp + (size_t)(t0 + TB) * D_, &sBT[cur ^ 1][DB], tid);

        // build Atilde (thread t owns row t); fold g into columns 1..16
        if (tid < TB) {
            int t = tid;
            float val = 1.0f;
            for (int tp = t; tp >= 0; --tp) {
                sA[t * KP + 1 + tp] = gp[t0 + tp] * val;
                val *= ap[t0 + tp];
            }
            sA[t * KP + 0] = val;                       // carry coefficient
            for (int kc = t + 2; kc < KP; ++kc) sA[t * KP + kc] = 0.0f;
        }
#if HAS_ASYNC_LDS
        if (more) { WAIT_ASYNC(4); }                    // prev group complete
        else      { WAIT_ASYNC(0); }
#endif
        __syncthreads();

        const float* bt = &sBT[cur][0];
        v8f cfrag[2];
#pragma unroll
        for (int ti = 0; ti < 2; ++ti) {
            int nBase = (wave + 8 * ti) * 16;
            int col   = nBase + row;
            v8f cacc = {};
#pragma unroll
            for (int ks = 0; ks < 5; ++ks) {            // branchless fragments
                int k0 = ks * 4 + half * 2;
                v2f afrag, bfrag;
                afrag.x = sA[row * KP + k0];
                afrag.y = sA[row * KP + k0 + 1];
                bfrag.x = bt[k0 * DB + col];
                bfrag.y = bt[(k0 + 1) * DB + col];
                cacc = __builtin_amdgcn_wmma_f32_16x16x4_f32(
                    false, afrag, false, bfrag, (short)0, cacc, false, false);
            }
            cfrag[ti] = cacc;
            // C/D layout: VGPR i -> row (i + 8*half)
#pragma unroll
            for (int i = 0; i < 8; ++i) {
                int t = i + 8 * half;
                op[(size_t)(t0 + t) * D_ + col] = cacc[i];
            }
        }
        __syncthreads();                                // LDS reads complete

        // y_out (t=15) -> carry row 0 of the NEXT buffer (rows 1..16 belong to
        // the in-flight async writes; disjoint addresses, published by the
        // barrier after the next iteration's staging)
        if (half == 1) {
#pragma unroll
            for (int ti = 0; ti < 2; ++ti)
                sBT[cur ^ 1][(wave + 8 * ti) * 16 + row] = cfrag[ti][7];
        }
        cur ^= 1;
    }
}

// ------------------------------------------------------------------ launch
extern "C" void kernel_launch(void* const* d_in, const int* in_sizes, int n_in,
                              void* d_out, int out_size, void* d_ws, size_t ws_size,
                              hipStream_t stream) {
    const float* x  = (const float*)d_in[0];   // (B, L, D) f32
    const float* pg = (const float*)d_in[1];   // (B, L, 1) f32
    const float* og = (const float*)d_in[2];   // (B, L, 1) f32
    float* out = (float*)d_out;                // (B, L, D) f32

    float* ws   = (float*)d_ws;
    float* a    = ws;                          // B_*L_
    float* g    = a + B_ * L_;                 // B_*L_
    float* Bfin = g + B_ * L_;                 // B_*NC*D_
    float* Atot = Bfin + B_ * NC * D_;         // B_*NC (padded to 256)
    float* Init = Atot + 256;                  // B_*NC*D_

    hipLaunchKernelGGL(k0_gates, dim3((B_ * L_ + 255) / 256), dim3(256), 0, stream,
                       pg, og, a, g);
    hipLaunchKernelGGL(k1_chunks, dim3(B_ * NC * (D_ / 256)), dim3(256), 0, stream,
                       x, a, g, Bfin, Atot);
    hipLaunchKernelGGL(k2_scan, dim3((B_ * D_ + 255) / 256), dim3(256), 0, stream,
                       Bfin, Atot, Init);
    hipLaunchKernelGGL(k3_out, dim3(B_ * NC * 2), dim3(256), 0, stream,
                       x, a, g, Init, out);
}